// NumericalRiskBudgeting_63436666962548
// MI455X (gfx1250) — compile-verified
//
#include <hip/hip_runtime.h>
#include <math.h>

// Risk budgeting via damped Newton on the KKT system, one workgroup per sample.
//  - Sigma = A^T A on the matrix pipe (V_WMMA_F32_16X16X4_F32, full f32).
//  - Blocked Cholesky: 16-wide panels on VALU, rank-16 trailing updates on WMMA.
//  - All LDS rows padded to stride 65 -> conflict-free column accesses.
//  - Deferred-division sweeps: 1 barrier per column in panel and both solves.
//  - Wave32 __shfl_xor reductions replace serial thread-0 loops.

#define N      64
#define S      65          // padded LDS row stride (64 banks -> conflict-free)
#define NITERS 25
#define BLOCK  128         // 4 waves (wave32)

typedef float v2f __attribute__((ext_vector_type(2)));
typedef float v4f __attribute__((ext_vector_type(4)));
typedef float v8f __attribute__((ext_vector_type(8)));

__device__ __forceinline__ float wave_sum(float v) {
    #pragma unroll
    for (int off = 16; off > 0; off >>= 1) v += __shfl_xor(v, off, 32);
    return v;
}
__device__ __forceinline__ float wave_min(float v) {
    #pragma unroll
    for (int off = 16; off > 0; off >>= 1) v = fminf(v, __shfl_xor(v, off, 32));
    return v;
}

__global__ __launch_bounds__(BLOCK)
void risk_budget_kernel(const float* __restrict__ Aglob,
                        const float* __restrict__ bglob,
                        float* __restrict__ out)
{
    __shared__ __align__(16) float sH[N * S];    // A (k-major), then H per iter
    __shared__ __align__(16) float sSig[N * S];  // Sigma = A^T A
    __shared__ float sw[N], sb[N], sg[N], sy1[N], sy2[N], sdw[N], sdg[16];
    __shared__ float sred[8];
    __shared__ float s_lam, s_h, s_dlam, s_alpha, s_bsum;

    const int tid  = threadIdx.x;
    const int samp = blockIdx.x;
    const int lane = tid & 31;
    const int wave = tid >> 5;
    const int l16  = lane & 15;
    const int lh   = lane >> 4;

    const float* A = Aglob + (size_t)samp * N * N;   // row-major: A[k][i]

    // ---- stage A (b128 coalesced reads) into padded LDS, and b ----
    for (int q4 = tid; q4 < (N * N) / 4; q4 += BLOCK) {
        const v4f v = *(const v4f*)(A + 4 * q4);
        const int r = q4 >> 4, c = (q4 & 15) * 4;
        sH[r * S + c + 0] = v.x;
        sH[r * S + c + 1] = v.y;
        sH[r * S + c + 2] = v.z;
        sH[r * S + c + 3] = v.w;
    }
    if (tid < N) sb[tid] = bglob[(size_t)samp * N + tid];
    __syncthreads();

    // ---- Sigma = A^T A via WMMA f32 16x16x4 ----
    // Wave owns tile-row I0; A-op (M=l16, K=2*lh+v) and B-op (K, N=l16) both
    // read column-slices of A: consecutive lanes -> consecutive banks.
    v8f acc[4];
    #pragma unroll
    for (int J = 0; J < 4; ++J) acc[J] = (v8f){0,0,0,0,0,0,0,0};
    const int I0 = wave;
    #pragma unroll
    for (int kb = 0; kb < 16; ++kb) {
        const int k0 = kb * 4 + 2 * lh;
        v2f av;
        av.x = sH[(k0 + 0) * S + I0 * 16 + l16];
        av.y = sH[(k0 + 1) * S + I0 * 16 + l16];
        #pragma unroll
        for (int J = 0; J < 4; ++J) {
            v2f bv;
            bv.x = sH[(k0 + 0) * S + J * 16 + l16];
            bv.y = sH[(k0 + 1) * S + J * 16 + l16];
            acc[J] = __builtin_amdgcn_wmma_f32_16x16x4_f32(
                false, av, false, bv, (short)0, acc[J], false, false);
        }
    }
    __syncthreads();   // done reading A from sH
    #pragma unroll
    for (int J = 0; J < 4; ++J)
        #pragma unroll
        for (int r = 0; r < 8; ++r)
            sSig[(I0 * 16 + r + 8 * lh) * S + J * 16 + l16] = acc[J][r];

    // ---- feasible interior start: w = b / sum(b), lam = 0 ----
    {
        float bp = (tid < N) ? sb[tid] : 0.f;
        bp = wave_sum(bp);
        if (lane == 0) sred[wave] = bp;
        __syncthreads();
        if (tid == 0) { s_bsum = sred[0] + sred[1] + sred[2] + sred[3]; s_lam = 0.f; }
        __syncthreads();
        if (tid < N) sw[tid] = sb[tid] / s_bsum;
        __syncthreads();
    }

    // ---- 25 damped Newton iterations + 1 exact (undamped) step ----
    for (int iter = 0; iter <= NITERS; ++iter) {
        const bool damped = (iter < NITERS);

        // g = Sigma w - b/w + lam ;  h = sum(w) - 1 (shuffle reduction)
        float wv = (tid < N) ? sw[tid] : 0.f;
        if (tid < N) {
            float accg = 0.f;
            for (int j = 0; j < N; ++j) accg += sSig[tid * S + j] * sw[j];
            sg[tid] = accg - sb[tid] / sw[tid] + s_lam;
        }
        {
            const float hp = wave_sum(wv);
            if (lane == 0) sred[wave] = hp;
        }
        __syncthreads();
        if (tid == 0) s_h = (sred[0] + sred[1] + sred[2] + sred[3]) - 1.0f;

        // H = Sigma + diag(b/w^2); rhs1 = g, rhs2 = ones  (b128 LDS copy)
        for (int q4 = tid; q4 < (N * S) / 4; q4 += BLOCK)
            *(v4f*)(sH + 4 * q4) = *(const v4f*)(sSig + 4 * q4);
        __syncthreads();
        if (tid < N) {
            sH[tid * S + tid] += sb[tid] / (sw[tid] * sw[tid]);
            sy1[tid] = sg[tid];
            sy2[tid] = 1.0f;
        }
        __syncthreads();

        // ---- blocked Cholesky: 16-wide panels + WMMA rank-16 trailing ----
        for (int bk = 0; bk < 4; ++bk) {
            const int c0 = bk * 16;
            // Panel columns c0..c0+15 over rows j..63; scaling deferred ->
            // one barrier per column (no writes to column j in its region).
            for (int jj = 0; jj < 16; ++jj) {
                const int j = c0 + jj;
                const float hjj  = sH[j * S + j];     // = d_j^2
                const float invd = 1.0f / hjj;
                if (tid == 0) sdg[jj] = sqrtf(hjj);
                const int rem = 15 - jj;              // panel cols right of j
                for (int p = tid; p < rem * N; p += BLOCK) {
                    const int kk = p >> 6;
                    const int i  = p & 63;
                    const int k  = j + 1 + kk;
                    if (i >= k)
                        sH[i * S + k] -= sH[i * S + j] * sH[k * S + j] * invd;
                }
                __syncthreads();
            }
            // scale panel: L[:,j] = col_j / d_j ; diag = d_j
            for (int p = tid; p < 16 * N; p += BLOCK) {
                const int jj = p >> 6;
                const int i  = p & 63;
                const int j  = c0 + jj;
                if (i > j)       sH[i * S + j] *= (1.0f / sdg[jj]);
                else if (i == j) sH[j * S + j]  = sdg[jj];
            }
            __syncthreads();
            // Trailing update on WMMA: H[I,J] -= L21_I * L21_J^T (16x16x16).
            // f32 WMMA has no A-negate, so negate the A operand at load.
            const int nb = 3 - bk;
            if (nb > 0) {
                const int npairs = (nb * (nb + 1)) >> 1;
                for (int t = wave; t < npairs; t += 4) {   // wave-uniform branch
                    int Ib = 0, Jb = 0, cnt = 0;
                    for (int jB = bk + 1; jB <= 3; ++jB)
                        for (int iB = jB; iB <= 3; ++iB) {
                            if (cnt == t) { Ib = iB; Jb = jB; }
                            ++cnt;
                        }
                    v8f cacc;
                    #pragma unroll
                    for (int r = 0; r < 8; ++r)
                        cacc[r] = sH[(Ib * 16 + r + 8 * lh) * S + Jb * 16 + l16];
                    #pragma unroll
                    for (int q = 0; q < 4; ++q) {
                        const int k = c0 + q * 4 + 2 * lh;
                        v2f av, bv;
                        av.x = -sH[(Ib * 16 + l16) * S + k];
                        av.y = -sH[(Ib * 16 + l16) * S + k + 1];
                        bv.x =  sH[(Jb * 16 + l16) * S + k];
                        bv.y =  sH[(Jb * 16 + l16) * S + k + 1];
                        cacc = __builtin_amdgcn_wmma_f32_16x16x4_f32(
                            false, av, false, bv, (short)0, cacc, false, false);
                    }
                    #pragma unroll
                    for (int r = 0; r < 8; ++r)
                        sH[(Ib * 16 + r + 8 * lh) * S + Jb * 16 + l16] = cacc[r];
                }
            }
            __syncthreads();
        }

        // ---- forward solve L y = rhs (2 RHS, deferred division) ----
        for (int c = 0; c < N; ++c) {
            const float invd = 1.0f / sH[c * S + c];
            const float y1c = sy1[c] * invd;
            const float y2c = sy2[c] * invd;
            if (tid > c && tid < N) {
                const float l = sH[tid * S + c];
                sy1[tid] -= l * y1c;
                sy2[tid] -= l * y2c;
            }
            __syncthreads();
        }
        if (tid < N) {
            const float invd = 1.0f / sH[tid * S + tid];
            sy1[tid] *= invd; sy2[tid] *= invd;
        }
        __syncthreads();
        // ---- backward solve L^T z = y (deferred division) ----
        for (int c = N - 1; c >= 0; --c) {
            const float invd = 1.0f / sH[c * S + c];
            const float z1c = sy1[c] * invd;
            const float z2c = sy2[c] * invd;
            if (tid < c) {
                const float l = sH[c * S + tid];   // L^T[r][c] = L[c][r]
                sy1[tid] -= l * z1c;
                sy2[tid] -= l * z2c;
            }
            __syncthreads();
        }
        if (tid < N) {
            const float invd = 1.0f / sH[tid * S + tid];
            sy1[tid] *= invd; sy2[tid] *= invd;
        }
        __syncthreads();

        // Schur complement: dlam = (h - 1^T z1)/(1^T z2); dw = -(z1 + dlam*z2)
        {
            float p1 = (tid < N) ? sy1[tid] : 0.f;
            float p2 = (tid < N) ? sy2[tid] : 0.f;
            p1 = wave_sum(p1);
            p2 = wave_sum(p2);
            if (lane == 0) { sred[wave] = p1; sred[4 + wave] = p2; }
        }
        __syncthreads();
        if (tid == 0) {
            const float a1 = sred[0] + sred[1] + sred[2] + sred[3];
            const float a2 = sred[4] + sred[5] + sred[6] + sred[7];
            s_dlam = (s_h - a1) / a2;
        }
        __syncthreads();
        if (tid < N) sdw[tid] = -(sy1[tid] + s_dlam * sy2[tid]);
        __syncthreads();

        // fraction-to-boundary damping (exact step on the final iteration)
        {
            float cand = __builtin_inff();
            if (damped && tid < N && sdw[tid] < 0.f) cand = -sw[tid] / sdw[tid];
            cand = wave_min(cand);
            if (lane == 0) sred[wave] = cand;
        }
        __syncthreads();
        if (tid == 0) {
            float alpha = 1.0f;
            if (damped) {
                const float step = fminf(fminf(sred[0], sred[1]),
                                         fminf(sred[2], sred[3]));
                alpha = fminf(1.0f, 0.99f * step);
            }
            s_alpha = alpha;
            s_lam  += alpha * s_dlam;
        }
        __syncthreads();
        if (tid < N) sw[tid] += s_alpha * sdw[tid];
        __syncthreads();
    }

    if (tid < N) out[(size_t)samp * N + tid] = sw[tid];
}

extern "C" void kernel_launch(void* const* d_in, const int* in_sizes, int n_in,
                              void* d_out, int out_size, void* d_ws, size_t ws_size,
                              hipStream_t stream) {
    const float* A   = (const float*)d_in[0];   // covmat_sqrt: (B, 64, 64) f32
    const float* b   = (const float*)d_in[1];   // b:           (B, 64)     f32
    float*       out = (float*)d_out;           // ws + dw:     (B, 64)     f32
    const int nsamp = in_sizes[1] / N;          // 8192
    risk_budget_kernel<<<nsamp, BLOCK, 0, stream>>>(A, b, out);
}